// SuperLoss_89575837926209
// MI455X (gfx1250) — compile-verified
//
#include <hip/hip_runtime.h>
#include <stdint.h>

#define B_ROWS      2048
#define C_COLS      32000
#define TPB         256
#define F4_PER_ROW  8000          // 32000 / 4
#define FULL_CHUNKS 31            // 31 * 256 = 7936 float4
#define TAIL_F4     64            // 8000 - 7936

#define TAU_F     10.37349118f    // ln(32000)
#define NEG2_E   -0.7357588823f   // -2/e
#define INV_T     0.25f           // 1/TEMP

typedef __attribute__((ext_vector_type(2))) float v2f;
typedef __attribute__((ext_vector_type(8))) float v8f;

// ---- gfx1250 async global->LDS helpers (inline asm; tracked by ASYNCcnt) ----
__device__ __forceinline__ unsigned lds_off_u32(const void* p) {
  // generic -> LDS addrspace(3); ptrtoint gives the byte offset from LDS_BASE
  return (unsigned)(unsigned long long)(__attribute__((address_space(3))) const char*)p;
}

__device__ __forceinline__ void async_b128(unsigned lds_dst, const void* sbase, unsigned byte_off) {
  // GVS mode: mem = SGPR64 base + VGPR32 offset ; dst = LDS addr in VGPR
  asm volatile("global_load_async_to_lds_b128 %0, %1, %2"
               :: "v"(lds_dst), "v"(byte_off), "s"(sbase)
               : "memory");
}

#define WAIT_ASYNC_2() asm volatile("s_wait_asynccnt 0x2" ::: "memory")
#define WAIT_ASYNC_0() asm volatile("s_wait_asynccnt 0x0" ::: "memory")
#define WAIT_DS_0()    asm volatile("s_wait_dscnt 0x0"    ::: "memory")

// ---- online softmax state update (student: base & /T ; teacher: /T + U) ----
__device__ __forceinline__ void upd(float l, float t,
                                    float& m,  float& s1, float& s2,
                                    float& mt, float& st, float& U) {
  if (l > m) {                       // rare: rescale
    float r = expf(m - l);           // exp(-inf)=0 on first hit
    s1 = s1 * r + 1.0f;
    s2 = s2 * sqrtf(sqrtf(r)) + 1.0f;
    m = l;
  } else {
    float e = expf(l - m);
    s1 += e;
    s2 += sqrtf(sqrtf(e));           // e^{(l-m)/4}
  }
  if (t > mt) {
    float r = expf((mt - t) * INV_T);
    st = st * r + 1.0f;
    U  = U * r + (t - l) * INV_T;
    mt = t;
  } else {
    float e = expf((t - mt) * INV_T);
    st += e;
    U   = fmaf(e, (t - l) * INV_T, U);
  }
}

// merge two online-softmax states
__device__ __forceinline__ void merge(float& m,  float& s1, float& s2,
                                      float& mt, float& st, float& U,
                                      float mb, float s1b, float s2b,
                                      float mtb, float stb, float Ub) {
  float M  = fmaxf(m, mb);
  float ea = expf(m - M), eb = expf(mb - M);
  s1 = s1 * ea + s1b * eb;
  s2 = s2 * sqrtf(sqrtf(ea)) + s2b * sqrtf(sqrtf(eb));
  m  = M;
  float Mt = fmaxf(mt, mtb);
  float fa = expf((mt - Mt) * INV_T), fb = expf((mtb - Mt) * INV_T);
  st = st * fa + stb * fb;
  U  = U  * fa + Ub  * fb;
  mt = Mt;
}

// Lambert W, mirrors the reference's init + 8 Halley iterations exactly
__device__ __forceinline__ float lambertw_f(float y) {
  float ey1     = fmaxf(2.718281828459045f * y + 1.0f, 0.0f);
  float w_small = -1.0f + sqrtf(2.0f * ey1);
  float ys      = fmaxf(y, 1.0f + 1e-6f);
  float ly      = logf(ys);
  float w_big   = ly - logf(fmaxf(ly, 1e-6f));
  float w       = (y < 1.0f) ? w_small : w_big;
#pragma unroll
  for (int i = 0; i < 8; ++i) {
    float ew    = expf(w);
    float f     = w * ew - y;
    float wp1   = w + 1.0f;
    float safe  = (fabsf(wp1) > 1e-6f) ? wp1 : 1e-6f;
    float denom = ew * wp1 - (w + 2.0f) * f / (2.0f * safe);
    denom       = (fabsf(denom) > 1e-12f) ? denom : 1e-12f;
    w           = w - f / denom;
  }
  return w;
}

// ---- kernel 1: one block per row, async double-buffered streaming ----
__global__ void __launch_bounds__(TPB) superloss_rows(
    const float* __restrict__ logits, const float* __restrict__ teacher,
    const int* __restrict__ targets,  const float* __restrict__ ones2,
    float* __restrict__ row_loss) {
  __shared__ float4 sL[2][TPB];
  __shared__ float4 sT[2][TPB];
  __shared__ float  red[8][6];

  const int tid = threadIdx.x;
  const int row = blockIdx.x;
  const float* lrow = logits  + (size_t)row * C_COLS;
  const float* trow = teacher + (size_t)row * C_COLS;

  float m = -INFINITY, s1 = 0.f, s2 = 0.f;
  float mt = -INFINITY, st = 0.f, U = 0.f;

  // prologue: chunks 0 and 1 in flight (2 async ops each)
  {
    unsigned off0 = (unsigned)(tid * 16);
    async_b128(lds_off_u32(&sL[0][tid]), lrow, off0);
    async_b128(lds_off_u32(&sT[0][tid]), trow, off0);
    unsigned off1 = (unsigned)((TPB + tid) * 16);
    async_b128(lds_off_u32(&sL[1][tid]), lrow, off1);
    async_b128(lds_off_u32(&sT[1][tid]), trow, off1);
  }

  for (int c = 0; c < FULL_CHUNKS; ++c) {
    if (c < FULL_CHUNKS - 1) { WAIT_ASYNC_2(); } else { WAIT_ASYNC_0(); }
    float4 lv = sL[c & 1][tid];
    float4 tv = sT[c & 1][tid];
    upd(lv.x, tv.x, m, s1, s2, mt, st, U);
    upd(lv.y, tv.y, m, s1, s2, mt, st, U);
    upd(lv.z, tv.z, m, s1, s2, mt, st, U);
    upd(lv.w, tv.w, m, s1, s2, mt, st, U);
    WAIT_DS_0();                         // LDS reads retired before buffer reuse
    if (c + 2 < FULL_CHUNKS) {
      unsigned off = (unsigned)(((c + 2) * TPB + tid) * 16);
      async_b128(lds_off_u32(&sL[c & 1][tid]), lrow, off);
      async_b128(lds_off_u32(&sT[c & 1][tid]), trow, off);
    }
  }

  // tail: 64 remaining float4 via direct loads
  if (tid < TAIL_F4) {
    const float4* l4 = (const float4*)lrow;
    const float4* t4 = (const float4*)trow;
    float4 lv = l4[FULL_CHUNKS * TPB + tid];
    float4 tv = t4[FULL_CHUNKS * TPB + tid];
    upd(lv.x, tv.x, m, s1, s2, mt, st, U);
    upd(lv.y, tv.y, m, s1, s2, mt, st, U);
    upd(lv.z, tv.z, m, s1, s2, mt, st, U);
    upd(lv.w, tv.w, m, s1, s2, mt, st, U);
  }

  // wave32 reduction
#pragma unroll
  for (int o = 16; o >= 1; o >>= 1) {
    float mb  = __shfl_xor(m,  o, 32);
    float s1b = __shfl_xor(s1, o, 32);
    float s2b = __shfl_xor(s2, o, 32);
    float mtb = __shfl_xor(mt, o, 32);
    float stb = __shfl_xor(st, o, 32);
    float Ub  = __shfl_xor(U,  o, 32);
    merge(m, s1, s2, mt, st, U, mb, s1b, s2b, mtb, stb, Ub);
  }
  if ((tid & 31) == 0) {
    int w = tid >> 5;
    red[w][0] = m;  red[w][1] = s1; red[w][2] = s2;
    red[w][3] = mt; red[w][4] = st; red[w][5] = U;
  }
  __syncthreads();

  if (tid == 0) {
    float M = red[0][0], S1 = red[0][1], S2 = red[0][2];
    float Mt = red[0][3], St = red[0][4], Uu = red[0][5];
    for (int w = 1; w < 8; ++w)
      merge(M, S1, S2, Mt, St, Uu,
            red[w][0], red[w][1], red[w][2], red[w][3], red[w][4], red[w][5]);

    float l_tgt = lrow[targets[row]];
    float logZ  = M + logf(S1);
    float ce    = logZ - l_tgt;
    float logZT = M * INV_T + logf(S2);
    float kl    = Uu / St - Mt * INV_T - logf(St) + logZT;

    float o0 = ones2[row * 2 + 0], o1 = ones2[row * 2 + 1];
    float base = o0 * ce + 16.0f * o1 * kl;

    float y     = 0.5f * fmaxf(NEG2_E, base - TAU_F);  // LAM = 1
    float w     = lambertw_f(y);
    float sigma = expf(-w);
    float lsig  = logf(sigma);
    row_loss[row] = (base - TAU_F) * sigma + lsig * lsig;
  }
}

// ---- kernel 2: exact-f32 batch sum via v_wmma_f32_16x16x4_f32 (A = ones) ----
__global__ void __launch_bounds__(32) finalize_wmma(const float* __restrict__ row_loss,
                                                    float* __restrict__ out) {
  const int lane = threadIdx.x;
  v2f a; a[0] = 1.0f; a[1] = 1.0f;       // A = all-ones 16x4
  v8f c = {};
  const v2f* p = (const v2f*)row_loss;   // 2048 floats = 1024 v2f
#pragma unroll 4
  for (int i = 0; i < 32; ++i) {
    v2f b = p[i * 32 + lane];            // 64 values per WMMA (B = 4x16 f32)
    // D[i][j] += sum_k B[k][j]  (exact f32 FMA accumulation)
    c = __builtin_amdgcn_wmma_f32_16x16x4_f32(false, a, false, b, (short)0, c, false, false);
  }
  // c[0], lanes 0..15 hold D[0][0..15] = column sums; reduce 16 columns
  float v = c[0];
  v += __shfl_xor(v, 1, 32);
  v += __shfl_xor(v, 2, 32);
  v += __shfl_xor(v, 4, 32);
  v += __shfl_xor(v, 8, 32);
  if (lane == 0) out[0] = v * (1.0f / 2048.0f);
}

extern "C" void kernel_launch(void* const* d_in, const int* in_sizes, int n_in,
                              void* d_out, int out_size, void* d_ws, size_t ws_size,
                              hipStream_t stream) {
  (void)in_sizes; (void)n_in; (void)out_size; (void)ws_size;
  const float* logits  = (const float*)d_in[0];
  const float* teacher = (const float*)d_in[1];
  const int*   targets = (const int*)d_in[2];
  const float* ones2   = (const float*)d_in[3];
  float* out = (float*)d_out;
  float* ws  = (float*)d_ws;             // 2048 per-row losses

  superloss_rows<<<B_ROWS, TPB, 0, stream>>>(logits, teacher, targets, ones2, ws);
  finalize_wmma<<<1, 32, 0, stream>>>(ws, out);
}